// WeightMaskGenerator_74947179315822
// MI455X (gfx1250) — compile-verified
//
#include <hip/hip_runtime.h>

// Problem constants: B=8, C=19, H=W=512
#define WID 512
#define HW  (512 * 512)
#define NB  8
#define NC  19
#define NPIX (NB * HW)

typedef __attribute__((ext_vector_type(2))) float v2f;
typedef __attribute__((ext_vector_type(8))) float v8f;

__device__ __forceinline__ v8f wmma_f32(v2f a, v2f b, v8f c) {
  // V_WMMA_F32_16X16X4_F32: D(16x16,f32) = A(16x4,f32) x B(4x16,f32) + C
  return __builtin_amdgcn_wmma_f32_16x16x4_f32(
      /*neg_a=*/false, a, /*neg_b=*/false, b,
      /*c_mod=*/(short)0, c, /*reuse_a=*/false, /*reuse_b=*/false);
}

// jnp.pad(..., mode='reflect') index mapping (mirror without edge repeat).
__device__ __forceinline__ int reflect_idx(int i, int n) {
  i = (i < 0) ? -i : i;                 // branchless: v_max_i32
  return (i >= n) ? (2 * n - 2 - i) : i;
}

// ---------------------------------------------------------------------------
// K1: fused channel einsum  mask[b,h,w] = sum_c mp[b,c,h,w]*aw[c]
//     (single full read of the 160MB input) + zero the per-image max slots.
// ---------------------------------------------------------------------------
__global__ void prep_kernel(const float* __restrict__ mp,
                            const float* __restrict__ aw,
                            float* __restrict__ einsum,
                            float* __restrict__ maxbuf) {
  if (blockIdx.x == 0 && threadIdx.x < 16) maxbuf[threadIdx.x] = 0.0f;
  long p = (long)blockIdx.x * blockDim.x + threadIdx.x;  // p < NPIX exactly
  int b = (int)(p / HW);
  long rem = p - (long)b * HW;
  const float* base = mp + (long)b * NC * HW + rem;
  float acc = 0.0f;
#pragma unroll
  for (int c = 0; c < NC; ++c) acc += base[(long)c * HW] * aw[c];
  einsum[p] = acc;
}

// ---------------------------------------------------------------------------
// Builds s_g: zero-padded, PRE-NORMALIZED Gaussian table.
// s_g[15 + d] = g[d]/sum(g) for d in [0,KSIZE), 0 elsewhere.
// Indexed as s_g[15 + (j - lidx)], always in bounds for j in [0, KPAD).
// Zero entries replace band-edge branches -> no EXEC divergence, and the
// normalization is baked in -> no per-WMMA-step scaling ALU.
// ---------------------------------------------------------------------------
template <int KSIZE, int GPAD>
__device__ __forceinline__ void build_taps(float* s_g, float* s_raw,
                                           float sigma) {
  for (int i = threadIdx.x; i < GPAD; i += blockDim.x) s_g[i] = 0.0f;
  __syncthreads();
  if (threadIdx.x < KSIZE) {
    float x = (float)threadIdx.x - (KSIZE - 1) * 0.5f;
    s_raw[threadIdx.x] = __expf(-0.5f * (x / sigma) * (x / sigma));
  }
  __syncthreads();
  if (threadIdx.x < KSIZE) {
    float ts = 0.0f;
    for (int k = 0; k < KSIZE; ++k) ts += s_raw[k];
    s_g[15 + threadIdx.x] = s_raw[threadIdx.x] / ts;
  }
  __syncthreads();
}

// ---------------------------------------------------------------------------
// Vertical (along H) Gaussian blur as banded WMMA GEMM:
//   D(16x16) = G(16xKSPAN) x In(KSPANx16),  G[m,j] = g[j-m]
// One wave per 16x16 output tile; data (B-operand) loads fully coalesced.
// eyes_mode=1 reads channels 4/5 of mask_parse directly (img = ch*8 + b).
// ---------------------------------------------------------------------------
template <int KSIZE>
__global__ void blur_v_kernel(const float* __restrict__ in,
                              float* __restrict__ out,
                              float sigma, int eyes_mode) {
  constexpr int R = KSIZE / 2;
  constexpr int KSPAN = 16 + 2 * R;
  constexpr int NK = (KSPAN + 3) / 4;
  constexpr int KPAD = NK * 4;
  constexpr int GPAD = KPAD + 16;      // index range: 15 + [-15, KPAD-1]

  __shared__ float s_g[GPAD];
  __shared__ float s_raw[KSIZE];
  build_taps<KSIZE, GPAD>(s_g, s_raw, sigma);

  const int lane = threadIdx.x & 31;
  const int wave = threadIdx.x >> 5;
  const int gt = blockIdx.x * (blockDim.x >> 5) + wave;
  const int img = gt >> 10;            // 32x32 tiles per 512x512 image
  const int y0 = (((gt >> 5) & 31) << 4);
  const int x0 = ((gt & 31) << 4);

  long ibase;
  if (eyes_mode) {
    int b = img & 7, ch = img >> 3;    // ch0 -> channel 4, ch1 -> channel 5
    ibase = ((long)b * NC + 4 + ch) * HW;
  } else {
    ibase = (long)img * HW;
  }
  const float* src = in + ibase;

  const int lidx = lane & 15;          // A: M index
  const int koff = (lane >> 4) * 2;    // K split across lane halves
  const int xb = x0 + lidx;
  const float* gg = s_g + 15 - lidx;   // per-lane normalized band base

  v8f acc = {0.f, 0.f, 0.f, 0.f, 0.f, 0.f, 0.f, 0.f};
#pragma unroll
  for (int kc = 0; kc < NK; ++kc) {
    const int j0 = kc * 4 + koff;
    v2f av;                            // consecutive pair -> ds_load_2addr
    av.x = gg[j0];
    av.y = gg[j0 + 1];
    const int r0 = reflect_idx(y0 - R + j0, 512);
    const int r1 = reflect_idx(y0 - R + j0 + 1, 512);
    v2f bv;
    bv.x = src[(long)r0 * WID + xb];   // 16 consecutive floats per half-wave
    bv.y = src[(long)r1 * WID + xb];
    acc = wmma_f32(av, bv, acc);
  }

  float* dst = out + (long)img * HW;
  const int mrow = (lane >> 4) * 8;    // D: lanes 16-31 hold rows M+8
#pragma unroll
  for (int v = 0; v < 8; ++v) dst[(long)(y0 + mrow + v) * WID + xb] = acc[v];
}

// ---------------------------------------------------------------------------
// Horizontal (along W) blur:  D(16x16) = A(16xKSPAN) x G(KSPANx16)
// A-tile staged in LDS (pitch = KPAD, conflict-free ds_load_2addr reads).
// MODE 0: plain.  MODE 1: * (1-eyes)(1-bg), per-image max via int atomicMax.
// MODE 2: write d_out + deterministic per-tile sums.
// ---------------------------------------------------------------------------
template <int KSIZE, int MODE>
__global__ void blur_h_kernel(const float* __restrict__ in,
                              float* __restrict__ out,
                              float sigma,
                              const float* __restrict__ mp,
                              float* __restrict__ maxbuf,
                              float* __restrict__ tilesum) {
  constexpr int R = KSIZE / 2;
  constexpr int KSPAN = 16 + 2 * R;
  constexpr int NK = (KSPAN + 3) / 4;
  constexpr int KPAD = NK * 4;
  constexpr int GPAD = KPAD + 16;

  __shared__ float s_g[GPAD];
  __shared__ float s_raw[KSIZE];
  __shared__ float s_a[4][16][KPAD];   // per-wave A tile

  const int lane = threadIdx.x & 31;
  const int wave = threadIdx.x >> 5;
  const int gt = blockIdx.x * (blockDim.x >> 5) + wave;
  const int img = gt >> 10;
  const int y0 = (((gt >> 5) & 31) << 4);
  const int x0 = ((gt & 31) << 4);
  const float* src = in + (long)img * HW;

  // Stage 16 rows x KPAD columns (reflect-padded) into LDS, coalesced.
  // (Done before the tap barriers so the barriers cover it too.)
  for (int idx = lane; idx < 16 * KPAD; idx += 32) {
    const int r = idx / KPAD;
    const int j = idx - r * KPAD;
    const int col = reflect_idx(x0 - R + j, 512);
    s_a[wave][r][j] = src[(long)(y0 + r) * WID + col];
  }
  build_taps<KSIZE, GPAD>(s_g, s_raw, sigma);  // includes __syncthreads

  const int lidx = lane & 15;          // A: M index / G: N index
  const int koff = (lane >> 4) * 2;
  const float* gg = s_g + 15 - lidx;
  const float* aa = &s_a[wave][lidx][0];

  v8f acc = {0.f, 0.f, 0.f, 0.f, 0.f, 0.f, 0.f, 0.f};
#pragma unroll
  for (int kc = 0; kc < NK; ++kc) {
    const int j0 = kc * 4 + koff;
    v2f av;                            // A from LDS (pair -> ds_load_2addr)
    av.x = aa[j0];
    av.y = aa[j0 + 1];
    v2f bv;                            // pre-normalized banded Gaussian
    bv.x = gg[j0];
    bv.y = gg[j0 + 1];
    acc = wmma_f32(av, bv, acc);
  }

  const int mrow = (lane >> 4) * 8;
  const int xb = x0 + lidx;
  float* dst = out + (long)img * HW;

  if (MODE == 1) {
    const int b = img & 7, ch = img >> 3;
    const float* eyesrc = mp + ((long)b * NC + 4 + ch) * HW;
    const float* bgsrc  = mp + (long)b * NC * HW;
    float lmax = 0.0f;
#pragma unroll
    for (int v = 0; v < 8; ++v) {
      const long off = (long)(y0 + mrow + v) * WID + xb;
      const float f = (1.0f - eyesrc[off]) * (1.0f - bgsrc[off]);
      const float val = acc[v] * f;
      dst[off] = val;
      lmax = fmaxf(lmax, val);
    }
    for (int s = 16; s; s >>= 1) lmax = fmaxf(lmax, __shfl_xor(lmax, s, 32));
    if (lane == 0)                      // nonneg floats: int compare == float
      atomicMax(reinterpret_cast<int*>(&maxbuf[img]), __float_as_int(lmax));
  } else if (MODE == 2) {
    float lsum = 0.0f;
#pragma unroll
    for (int v = 0; v < 8; ++v) {
      const long off = (long)(y0 + mrow + v) * WID + xb;
      const float val = acc[v];
      dst[off] = val;
      lsum += val;
    }
    for (int s = 16; s; s >>= 1) lsum += __shfl_xor(lsum, s, 32);
    if (lane == 0) tilesum[gt] = lsum;  // deterministic: no atomics
  } else {
#pragma unroll
    for (int v = 0; v < 8; ++v)
      dst[(long)(y0 + mrow + v) * WID + xb] = acc[v];
  }
}

// ---------------------------------------------------------------------------
// K4: mask = einsum + (w_l/max_l + w_r/max_r) * eye_shadows_weight  (in place)
// ---------------------------------------------------------------------------
__global__ void combine_kernel(float* __restrict__ mask,
                               const float* __restrict__ wbuf,
                               const float* __restrict__ maxbuf,
                               const float* __restrict__ esw_p) {
  long p = (long)blockIdx.x * blockDim.x + threadIdx.x;
  const int b = (int)(p / HW);
  const long rem = p - (long)b * HW;
  const float esw = esw_p[0];
  const float wl = wbuf[(long)b * HW + rem];        // img = b      (channel 4)
  const float wr = wbuf[(long)(8 + b) * HW + rem];  // img = 8 + b  (channel 5)
  mask[p] += (wl / maxbuf[b] + wr / maxbuf[8 + b]) * esw;
}

// ---------------------------------------------------------------------------
// K7: fixed-order per-batch sum of tile sums -> store HW/sum (mean-norm scale)
// ---------------------------------------------------------------------------
__global__ void finalize_sums(const float* __restrict__ tilesum,
                              float* __restrict__ invbuf) {
  const int b = threadIdx.x;
  if (b < NB) {
    float s = 0.0f;
    for (int i = 0; i < 1024; ++i) s += tilesum[b * 1024 + i];
    invbuf[b] = (float)HW / s;
  }
}

// K8: out[p] *= HW / sum[b]
__global__ void scale_kernel(float* __restrict__ out,
                             const float* __restrict__ invbuf) {
  long p = (long)blockIdx.x * blockDim.x + threadIdx.x;
  out[p] *= invbuf[(int)(p / HW)];
}

// ---------------------------------------------------------------------------
extern "C" void kernel_launch(void* const* d_in, const int* in_sizes, int n_in,
                              void* d_out, int out_size, void* d_ws, size_t ws_size,
                              hipStream_t stream) {
  const float* mp  = (const float*)d_in[0];   // (8,19,512,512)
  const float* aw  = (const float*)d_in[1];   // (19,)
  const float* esw = (const float*)d_in[2];   // (1,)
  float* out = (float*)d_out;                 // (8,1,512,512)
  float* ws  = (float*)d_ws;

  // Workspace layout (floats):
  float* einsum = ws;                          // 8*HW   (also final mask, in-place)
  float* tmp    = einsum + (long)NB * HW;      // 16*HW  (blur-V outputs)
  float* wbuf   = tmp + (long)16 * HW;         // 16*HW  (eye weight masks)
  float* smalls = wbuf + (long)16 * HW;
  float* maxbuf  = smalls;                     // 16
  float* invbuf  = smalls + 16;                // 8
  float* tilesum = smalls + 32;                // 8192

  const int pix_blocks = NPIX / 256;           // 8192

  // K1: einsum over channels (single read of mask_parse), init max slots
  prep_kernel<<<pix_blocks, 256, 0, stream>>>(mp, aw, einsum, maxbuf);

  // K2: vertical 51-tap blur of eye channels (16 images), WMMA banded GEMM
  blur_v_kernel<51><<<16 * 1024 / 4, 128, 0, stream>>>(mp, tmp, 15.0f, /*eyes*/1);

  // K3: horizontal 51-tap blur + (1-eyes)(1-bg) factor + per-image max
  blur_h_kernel<51, 1><<<16 * 1024 / 4, 128, 0, stream>>>(
      tmp, wbuf, 15.0f, mp, maxbuf, nullptr);

  // K4: combine einsum + normalized eye masks * eye_shadows_weight
  combine_kernel<<<pix_blocks, 256, 0, stream>>>(einsum, wbuf, maxbuf, esw);

  // K5: vertical 21-tap blur of combined mask (8 images)
  blur_v_kernel<21><<<NB * 1024 / 4, 128, 0, stream>>>(einsum, tmp, 2.0f, /*eyes*/0);

  // K6: horizontal 21-tap blur -> d_out, deterministic per-tile sums
  blur_h_kernel<21, 2><<<NB * 1024 / 4, 128, 0, stream>>>(
      tmp, out, 2.0f, nullptr, nullptr, tilesum);

  // K7/K8: deterministic mean-norm
  finalize_sums<<<1, 32, 0, stream>>>(tilesum, invbuf);
  scale_kernel<<<pix_blocks, 256, 0, stream>>>(out, invbuf);
}